// SparseGCNLayer_47510928228756
// MI455X (gfx1250) — compile-verified
//
#include <hip/hip_runtime.h>

typedef float v2f __attribute__((ext_vector_type(2)));
typedef float v8f __attribute__((ext_vector_type(8)));

#define DIM 128

// ---------------- zero workspace (deg + AX are contiguous) ----------------
__global__ void gcn_zero_kernel(float* __restrict__ p, int n) {
    int i = blockIdx.x * blockDim.x + threadIdx.x;
    if (i < n) p[i] = 0.0f;
}

// ---------------- deg[i] = segment_sum(values, row) ----------------
__global__ void gcn_deg_kernel(const long long* __restrict__ row,
                               const float* __restrict__ vals,
                               float* __restrict__ deg, int E) {
    int e = blockIdx.x * blockDim.x + threadIdx.x;
    if (e < E) {
        atomicAdd(&deg[(int)row[e]], vals[e]);
    }
}

// ---------------- deg -> 1/sqrt(clip(deg, eps)) in place ----------------
__global__ void gcn_dis_kernel(float* __restrict__ deg, int n) {
    int i = blockIdx.x * blockDim.x + threadIdx.x;
    if (i < n) {
        float d = deg[i];
        d = d < 1e-9f ? 1e-9f : d;
        deg[i] = 1.0f / sqrtf(d);
    }
}

// ---------------- SpMM scatter: one wave per edge ----------------
// lane L handles columns [4L, 4L+4): a float4 of X[col], 4 f32 atomics to AX[row].
// X and AX (51.2 MB each) are L2-resident on MI455X (192 MB L2), so the
// gather + atomic scatter stays on-chip.
__global__ void __launch_bounds__(256)
gcn_spmm_kernel(const long long* __restrict__ row,
                const long long* __restrict__ col,
                const float* __restrict__ vals,
                const float* __restrict__ dis,
                const float* __restrict__ X,
                float* __restrict__ AX, int E) {
    int e    = blockIdx.x * (blockDim.x >> 5) + (threadIdx.x >> 5);
    int lane = threadIdx.x & 31;
    if (e >= E) return;
    int r = (int)row[e];
    int c = (int)col[e];
    float v = vals[e] * dis[r] * dis[c];
    const float4* xr = (const float4*)(X + (size_t)c * DIM);
    float4 x = xr[lane];                      // 128 floats = 32 float4 = 1 per lane
    float* out = AX + (size_t)r * DIM + lane * 4;
    atomicAdd(out + 0, v * x.x);
    atomicAdd(out + 1, v * x.y);
    atomicAdd(out + 2, v * x.z);
    atomicAdd(out + 3, v * x.w);
}

// ---------------- Out = relu(AX @ W^T) via V_WMMA_F32_16X16X4_F32 ----------------
// One wave computes a 16x16 tile of Out. Block = 256 threads = 8 waves = the 8
// column tiles of a 16-row slab. K-loop: 128/4 = 32 WMMAs, fully unrolled.
//
// Fragment layouts (ISA 7.12.2, f32):
//   A 16x4 : lanes 0-15 -> {A[m][k+0], A[m][k+1]}, lanes 16-31 -> {A[m][k+2], A[m][k+3]}
//   B 4x16 : lanes 0-15 -> {B[k+0][n], B[k+1][n]}, lanes 16-31 -> {B[k+2][n], B[k+3][n]}
//            with B[k][n] = W[col0+n][k]  (B = W^T tile)
//   C/D    : VGPR r, lanes 0-15 -> Out[row0+r][col0+n]; lanes 16-31 -> Out[row0+8+r][...]
__global__ void __launch_bounds__(256)
gcn_gemm_relu_kernel(const float* __restrict__ AX,
                     const float* __restrict__ W,
                     float* __restrict__ Out) {
    const int row0 = blockIdx.x * 16;
    const int wave = threadIdx.x >> 5;
    const int lane = threadIdx.x & 31;
    const int col0 = wave * 16;
    const int lsel = lane & 15;
    const int kh   = (lane >> 4) << 1;   // 0 for lanes 0-15, 2 for lanes 16-31

    const float* arow = AX + (size_t)(row0 + lsel) * DIM;  // A rows: AX[m][*]
    const float* brow = W  + (size_t)(col0 + lsel) * DIM;  // B cols: W[n][*]

    v8f c = {};
#pragma unroll
    for (int k = 0; k < DIM; k += 4) {
        v2f a = *(const v2f*)(arow + k + kh);   // 8B-aligned: global_load_b64
        v2f b = *(const v2f*)(brow + k + kh);
        c = __builtin_amdgcn_wmma_f32_16x16x4_f32(
                /*neg_a=*/false, a, /*neg_b=*/false, b,
                /*c_mod=*/(short)0, c, /*reuse_a=*/false, /*reuse_b=*/false);
    }

    const int mbase = row0 + ((lane >> 4) << 3);
    const int colw  = col0 + lsel;
#pragma unroll
    for (int r = 0; r < 8; ++r) {
        float v = c[r];
        Out[(size_t)(mbase + r) * DIM + colw] = v > 0.0f ? v : 0.0f;
    }
}

extern "C" void kernel_launch(void* const* d_in, const int* in_sizes, int n_in,
                              void* d_out, int out_size, void* d_ws, size_t ws_size,
                              hipStream_t stream) {
    const float*     X    = (const float*)d_in[0];
    const long long* ei   = (const long long*)d_in[1];   // int64 COO indices
    const float*     vals = (const float*)d_in[2];
    const float*     W    = (const float*)d_in[3];

    const int N = in_sizes[0] / DIM;   // 100000
    const int E = in_sizes[1] / 2;     // 1600000

    const long long* row = ei;
    const long long* col = ei + E;

    float* deg = (float*)d_ws;         // N floats (becomes d_inv_sqrt in place)
    float* AX  = deg + N;              // N*128 floats
    float* Out = (float*)d_out;

    // 1) zero deg + AX (contiguous: N*(1+128) floats)
    {
        int ztotal = N * (DIM + 1);
        int blocks = (ztotal + 255) / 256;
        gcn_zero_kernel<<<blocks, 256, 0, stream>>>(deg, ztotal);
    }
    // 2) degree scatter-add
    gcn_deg_kernel<<<(E + 255) / 256, 256, 0, stream>>>(row, vals, deg, E);
    // 3) d^{-1/2}
    gcn_dis_kernel<<<(N + 255) / 256, 256, 0, stream>>>(deg, N);
    // 4) SpMM: 1 wave per edge, 8 waves per block
    gcn_spmm_kernel<<<(E + 7) / 8, 256, 0, stream>>>(row, col, vals, deg, X, AX, E);
    // 5) WMMA GEMM + ReLU (N = 6250 * 16 exactly)
    gcn_gemm_relu_kernel<<<N / 16, 256, 0, stream>>>(AX, W, Out);
}